// ResGCNNet_25658134626484
// MI455X (gfx1250) — compile-verified
//
#include <hip/hip_runtime.h>
#include <hip/hip_bf16.h>

// ResGCN on gfx1250: fp32 WMMA GEMMs + L2-resident scatter-add aggregation.
// N=100000 nodes, 128 features, E=1.6M edges; whole working set fits MI455X's
// 192MB L2, so the edge scatter (hardware global_atomic_add_f32) runs at L2 bw.
// GEMM epilogue fuses the self-loop init (agg = h * dinv^2), saving a full
// read+write pass over the 51MB activation tensor per layer.

#define FEAT 128
#define OUTC 16
#define GN_EPS 1e-5f
#define STAT_ROWS 256

typedef float v2f __attribute__((ext_vector_type(2)));
typedef float v8f __attribute__((ext_vector_type(8)));

__device__ __forceinline__ void atomAddF32(float* p, float v) {
  // lowers to global_atomic_add_f32 (no return), tracked with STOREcnt
  __hip_atomic_fetch_add(p, v, __ATOMIC_RELAXED, __HIP_MEMORY_SCOPE_AGENT);
}

// ---------------------------------------------------------------- utilities
__global__ void zero_u32_kernel(unsigned int* __restrict__ p, long n) {
  long i = (long)blockIdx.x * blockDim.x + threadIdx.x;
  if (i < n) p[i] = 0u;
}

__global__ void count_kernel(const long long* __restrict__ dst,
                             unsigned int* __restrict__ cnt, int E) {
  int e = blockIdx.x * 256 + threadIdx.x;
  if (e < E)
    __hip_atomic_fetch_add(&cnt[dst[e]], 1u, __ATOMIC_RELAXED,
                           __HIP_MEMORY_SCOPE_AGENT);
}

// in-place: uint in-degree -> float rsqrt(deg + 1)  (self-loop included)
__global__ void dinv_kernel(unsigned int* __restrict__ buf, int n) {
  int i = blockIdx.x * 256 + threadIdx.x;
  if (i < n) {
    float d = (float)buf[i] + 1.0f;
    ((float*)buf)[i] = rsqrtf(d);
  }
}

// ------------------------------------------------- 128x128 GEMM (H = X * W)
// block = 256 thr = 8 waves; block computes 32 rows x 128 cols.
// Each wave owns one 16-col tile; A staged via LDS (pad 132 to avoid 16-way
// bank conflicts on the 512B row stride); B streamed from L2-resident W.
// Epilogue writes H and the self-loop-initialized aggregate A = H * dinv^2.
__global__ void __launch_bounds__(256)
gemm128_kernel(const float* __restrict__ X, const float* __restrict__ W,
               float* __restrict__ H, float* __restrict__ A,
               const float* __restrict__ dinv, int nrows) {
  __shared__ float As[32 * 132];
  const int t = threadIdx.x;
  const int w = t >> 5;
  const int l = t & 31;
  const int rowBase = blockIdx.x * 32;
  const bool full = (rowBase + 32) <= nrows;

  // cooperative 32x128 tile load (4 float4 per thread)
  const float4* Xg = (const float4*)(X + (long)rowBase * FEAT);
#pragma unroll
  for (int i = 0; i < 4; ++i) {
    int f4 = t + i * 256;          // 1024 float4 total
    int r = f4 >> 5;               // (f4*4)/128
    int c = (f4 & 31) * 4;
    float4 v;
    if (full || rowBase + r < nrows) v = Xg[f4];
    else { v.x = v.y = v.z = v.w = 0.f; }
    *(float4*)&As[r * 132 + c] = v;
  }
  __syncthreads();

  const int m = l & 15;            // lane's A row / B col within tile
  const int khalf = (l >> 4) << 1; // K sub-offset per lane half
  const int n = w * 16 + m;        // global output column
  v8f c0 = {};
  v8f c1 = {};
#pragma unroll 4
  for (int ks = 0; ks < 32; ++ks) {
    int k = ks * 4 + khalf;
    v2f a0 = *(const v2f*)&As[m * 132 + k];
    v2f a1 = *(const v2f*)&As[(m + 16) * 132 + k];
    v2f b;
    b.x = W[k * FEAT + n];
    b.y = W[(k + 1) * FEAT + n];
    c0 = __builtin_amdgcn_wmma_f32_16x16x4_f32(false, a0, false, b, (short)0,
                                               c0, false, false);
    c1 = __builtin_amdgcn_wmma_f32_16x16x4_f32(false, a1, false, b, (short)0,
                                               c1, false, false);
  }
  const int rowOff = (l >> 4) << 3;
  if (full) {
    // straight-line stores, no per-row exec masking
#pragma unroll
    for (int r = 0; r < 8; ++r) {
      int r0 = rowBase + rowOff + r;
      int r1 = r0 + 16;
      float s0 = dinv[r0]; s0 *= s0;       // self-loop norm = 1/deg
      float s1 = dinv[r1]; s1 *= s1;
      H[(long)r0 * FEAT + n] = c0[r];
      H[(long)r1 * FEAT + n] = c1[r];
      A[(long)r0 * FEAT + n] = c0[r] * s0;
      A[(long)r1 * FEAT + n] = c1[r] * s1;
    }
  } else {
#pragma unroll
    for (int r = 0; r < 8; ++r) {
      int r0 = rowBase + rowOff + r;
      int r1 = r0 + 16;
      if (r0 < nrows) {
        float s0 = dinv[r0]; s0 *= s0;
        H[(long)r0 * FEAT + n] = c0[r];
        A[(long)r0 * FEAT + n] = c0[r] * s0;
      }
      if (r1 < nrows) {
        float s1 = dinv[r1]; s1 *= s1;
        H[(long)r1 * FEAT + n] = c1[r];
        A[(long)r1 * FEAT + n] = c1[r] * s1;
      }
    }
  }
}

// ------------------------------------------------------------- edge scatter
// 8 threads per edge, 16 floats each: A[dst] += dinv[src]*dinv[dst]*H[src]
__global__ void scatter_kernel(const long long* __restrict__ src,
                               const long long* __restrict__ dst,
                               const float* __restrict__ dinv,
                               const float* __restrict__ H,
                               float* __restrict__ A, int E) {
  long tid = (long)blockIdx.x * 256 + threadIdx.x;
  if (tid >= (long)E * 8) return;
  int e = (int)(tid >> 3);
  int seg = (int)(tid & 7) * 16;
  long long s = src[e];
  long long d = dst[e];
  float nrm = dinv[s] * dinv[d];
  const float4* hs = (const float4*)(H + (long)s * FEAT + seg);
  float* od = A + (long)d * FEAT + seg;
#pragma unroll
  for (int j = 0; j < 4; ++j) {
    float4 v = hs[j];
    atomAddF32(od + j * 4 + 0, nrm * v.x);
    atomAddF32(od + j * 4 + 1, nrm * v.y);
    atomAddF32(od + j * 4 + 2, nrm * v.z);
    atomAddF32(od + j * 4 + 3, nrm * v.w);
  }
}

// ------------------------------------------- GraphNorm: one-pass column stats
// S[0:128] = sum(X), S[128:256] = sum(X^2)
__global__ void __launch_bounds__(256)
stats_kernel(const float* __restrict__ X, float* __restrict__ S, int nrows) {
  __shared__ float sh1[128];
  __shared__ float sh2[128];
  int c = threadIdx.x & 127;
  int half = threadIdx.x >> 7;
  long r0 = (long)blockIdx.x * STAT_ROWS + half;
  float s1 = 0.f, s2 = 0.f;
#pragma unroll 4
  for (int i = 0; i < STAT_ROWS; i += 2) {
    long r = r0 + i;
    if (r < nrows) {
      float v = X[r * FEAT + c];
      s1 += v;
      s2 += v * v;
    }
  }
  if (half) { sh1[c] = s1; sh2[c] = s2; }
  __syncthreads();
  if (!half) {
    s1 += sh1[c];
    s2 += sh2[c];
    atomAddF32(&S[c], s1);
    atomAddF32(&S[128 + c], s2);
  }
}

// fold conv-bias cb and GraphNorm(w,b,a) into per-channel y = relu(A*x + B)
__global__ void finstats_kernel(const float* __restrict__ S,
                                const float* __restrict__ cb,
                                const float* __restrict__ gw,
                                const float* __restrict__ gb,
                                const float* __restrict__ ga,
                                float* __restrict__ AB, float invN) {
  int c = threadIdx.x;  // 128 threads
  float m = S[c] * invN;                 // mean of raw aggregate
  float b = cb[c];
  float mb = m + b;                      // mean after bias
  float e2 = S[128 + c] * invN + 2.f * b * m + b * b;  // E[(x+b)^2]
  float a = ga[c];
  float var = e2 - mb * mb * (2.f * a - a * a);        // E[(x'-a*m)^2]
  float rstd = rsqrtf(var + GN_EPS);
  float sc = gw[c] * rstd;
  AB[c] = sc;
  AB[128 + c] = sc * (b - a * mb) + gb[c];
}

__global__ void transform_kernel(const float* __restrict__ X,
                                 const float* __restrict__ AB,
                                 float* __restrict__ Y, long n4) {
  long i = (long)blockIdx.x * 256 + threadIdx.x;
  if (i >= n4) return;
  int c4 = (int)(i & 31);
  float4 v = ((const float4*)X)[i];
  float4 a = ((const float4*)AB)[c4];
  float4 b = ((const float4*)(AB + 128))[c4];
  v.x = fmaxf(fmaf(a.x, v.x, b.x), 0.f);
  v.y = fmaxf(fmaf(a.y, v.y, b.y), 0.f);
  v.z = fmaxf(fmaf(a.z, v.z, b.z), 0.f);
  v.w = fmaxf(fmaf(a.w, v.w, b.w), 0.f);
  ((float4*)Y)[i] = v;
}

// --------------------------- head: out = concat(x, x2) @ Wh + bh  (K=256,N=16)
__global__ void __launch_bounds__(256)
head_kernel(const float* __restrict__ X, const float* __restrict__ X2,
            const float* __restrict__ Wh, const float* __restrict__ bh,
            float* __restrict__ Out, int nrows) {
  int t = threadIdx.x;
  int w = t >> 5;
  int l = t & 31;
  int rowBase = blockIdx.x * 128 + w * 16;
  int m = l & 15;
  int khalf = (l >> 4) << 1;
  int n = m;                          // single 16-wide column tile
  long rowA = rowBase + m;
  const bool full = (blockIdx.x * 128 + 128) <= nrows;
  bool inb = rowA < nrows;
  const float* xr = X + rowA * FEAT;
  const float* x2r = X2 + rowA * FEAT;
  v8f c = {};
  // K = 0..127 from x
#pragma unroll 4
  for (int ks = 0; ks < 32; ++ks) {
    int k = ks * 4 + khalf;
    v2f a;
    if (inb) a = *(const v2f*)(xr + k);
    else { a.x = 0.f; a.y = 0.f; }
    v2f b;
    b.x = Wh[k * OUTC + n];
    b.y = Wh[(k + 1) * OUTC + n];
    c = __builtin_amdgcn_wmma_f32_16x16x4_f32(false, a, false, b, (short)0, c,
                                              false, false);
  }
  // K = 128..255 from x2
#pragma unroll 4
  for (int ks = 0; ks < 32; ++ks) {
    int k = ks * 4 + khalf;
    v2f a;
    if (inb) a = *(const v2f*)(x2r + k);
    else { a.x = 0.f; a.y = 0.f; }
    v2f b;
    b.x = Wh[(k + FEAT) * OUTC + n];
    b.y = Wh[(k + FEAT + 1) * OUTC + n];
    c = __builtin_amdgcn_wmma_f32_16x16x4_f32(false, a, false, b, (short)0, c,
                                              false, false);
  }
  float bias = bh[n];
  int rowOff = (l >> 4) << 3;
  if (full) {
#pragma unroll
    for (int r = 0; r < 8; ++r) {
      int row = rowBase + rowOff + r;
      Out[(long)row * OUTC + n] = c[r] + bias;
    }
  } else {
#pragma unroll
    for (int r = 0; r < 8; ++r) {
      int row = rowBase + rowOff + r;
      if (row < nrows) Out[(long)row * OUTC + n] = c[r] + bias;
    }
  }
}

// ---------------------------------------------------------------- launcher
extern "C" void kernel_launch(void* const* d_in, const int* in_sizes, int n_in,
                              void* d_out, int out_size, void* d_ws,
                              size_t ws_size, hipStream_t stream) {
  (void)n_in; (void)out_size; (void)ws_size;
  const float* x   = (const float*)d_in[0];
  const long long* ei = (const long long*)d_in[1];  // int64 [2,E]
  const float* W1  = (const float*)d_in[2];
  const float* b1  = (const float*)d_in[3];
  const float* g1w = (const float*)d_in[4];
  const float* g1b = (const float*)d_in[5];
  const float* g1a = (const float*)d_in[6];
  const float* W2  = (const float*)d_in[7];
  const float* b2  = (const float*)d_in[8];
  const float* g2w = (const float*)d_in[9];
  const float* g2b = (const float*)d_in[10];
  const float* g2a = (const float*)d_in[11];
  const float* Wh  = (const float*)d_in[12];
  const float* bh  = (const float*)d_in[13];
  float* out = (float*)d_out;

  const int n = in_sizes[0] / FEAT;
  const int E = in_sizes[1] / 2;
  const long long* srcv = ei;
  const long long* dstv = ei + E;

  // workspace carve-out (≈154 MB): dinv | h | agg | x1 | stats | AB
  char* ws = (char*)d_ws;
  size_t off = 0;
  auto carve = [&](size_t bytes) -> void* {
    void* p = ws + off;
    off += (bytes + 255) & ~(size_t)255;
    return p;
  };
  float* dinv = (float*)carve((size_t)n * 4);
  float* h    = (float*)carve((size_t)n * FEAT * 4);
  float* agg  = (float*)carve((size_t)n * FEAT * 4);
  float* x1   = (float*)carve((size_t)n * FEAT * 4);
  float* S    = (float*)carve(256 * 4);
  float* AB   = (float*)carve(256 * 4);

  const long n4 = (long)n * (FEAT / 4);
  const float invN = 1.0f / (float)n;
  const int gElem  = (int)((n4 + 255) / 256);
  const int gGemm  = (n + 31) / 32;
  const int gStat  = (n + STAT_ROWS - 1) / STAT_ROWS;
  const int gScat  = (int)(((long)E * 8 + 255) / 256);
  const int gHead  = (n + 127) / 128;

  // degrees -> dinv (in-place uint count then rsqrt)
  zero_u32_kernel<<<(n + 255) / 256, 256, 0, stream>>>((unsigned int*)dinv, n);
  count_kernel<<<(E + 255) / 256, 256, 0, stream>>>(dstv, (unsigned int*)dinv, E);
  dinv_kernel<<<(n + 255) / 256, 256, 0, stream>>>((unsigned int*)dinv, n);

  // ---- layer 1 (gemm writes h and self-loop-initialized agg)
  gemm128_kernel<<<gGemm, 256, 0, stream>>>(x, W1, h, agg, dinv, n);
  scatter_kernel<<<gScat, 256, 0, stream>>>(srcv, dstv, dinv, h, agg, E);
  zero_u32_kernel<<<1, 256, 0, stream>>>((unsigned int*)S, 256);
  stats_kernel<<<gStat, 256, 0, stream>>>(agg, S, n);
  finstats_kernel<<<1, 128, 0, stream>>>(S, b1, g1w, g1b, g1a, AB, invN);
  transform_kernel<<<gElem, 256, 0, stream>>>(agg, AB, x1, n4);

  // ---- layer 2 (reuse h/agg; x2 lands back in x1 buffer)
  gemm128_kernel<<<gGemm, 256, 0, stream>>>(x1, W2, h, agg, dinv, n);
  scatter_kernel<<<gScat, 256, 0, stream>>>(srcv, dstv, dinv, h, agg, E);
  zero_u32_kernel<<<1, 256, 0, stream>>>((unsigned int*)S, 256);
  stats_kernel<<<gStat, 256, 0, stream>>>(agg, S, n);
  finstats_kernel<<<1, 128, 0, stream>>>(S, b2, g2w, g2b, g2a, AB, invN);
  transform_kernel<<<gElem, 256, 0, stream>>>(agg, AB, x1, n4);

  // ---- head
  head_kernel<<<gHead, 256, 0, stream>>>(x, x1, Wh, bh, out, n);
}